// C4AutoregressivePrintf_62380105007593
// MI455X (gfx1250) — compile-verified
//
#include <hip/hip_runtime.h>
#include <stdint.h>

#define BLOCK 256
#define MAXM  4096
#define OUTC  65   // OUTPUT_SIZE + 1

typedef unsigned int u32x4 __attribute__((ext_vector_type(4)));
typedef int          i32x4 __attribute__((ext_vector_type(4)));
typedef int          i32x8 __attribute__((ext_vector_type(8)));
typedef float        v2f   __attribute__((ext_vector_type(2)));
typedef float        v8f   __attribute__((ext_vector_type(8)));

#if defined(__gfx1250__) && __has_builtin(__builtin_amdgcn_wmma_f32_16x16x4_f32)
#define USE_WMMA_RED 1
#endif

// silu / silu_threshold exactly as the reference (SCALE = 20)
__device__ __forceinline__ float silu_f(float z) {
  return z / (1.0f + __expf(-z));
}
__device__ __forceinline__ float sthr(float x) {
  float d = 20.0f * x;
  return (silu_f(d + 10.0f) - silu_f(d - 10.0f)) * 0.05f;
}

// deterministic wave32 butterfly reduction (fallback path)
__device__ __forceinline__ float wave_sum32(float v) {
  v += __shfl_xor(v, 16, 32);
  v += __shfl_xor(v,  8, 32);
  v += __shfl_xor(v,  4, 32);
  v += __shfl_xor(v,  2, 32);
  v += __shfl_xor(v,  1, 32);
  return v;
}

__global__ __launch_bounds__(BLOCK)
void c4_print_kernel(const float* __restrict__ memory,
                     const int*   __restrict__ addr,
                     const int*   __restrict__ out_ptr,
                     float*       __restrict__ out,
                     int M) {
  __shared__ __align__(16) float rowbuf[MAXM];
  __shared__ __align__(16) float red[2 * BLOCK];   // partials for WMMA/tree reduction
  __shared__ float wred[16];
  __shared__ float sval;
  __shared__ float tokens[OUTC];

  const int row  = blockIdx.x;
  const int tid  = threadIdx.x;
  const int lane = tid & 31;
  const int wid  = tid >> 5;
  const float* src = memory + (size_t)row * (size_t)M;

  // ---- Stage the 16KiB row into LDS via the Tensor Data Mover ----
#if defined(__gfx1250__) && __has_builtin(__builtin_amdgcn_tensor_load_to_lds) && __has_builtin(__builtin_amdgcn_s_wait_tensorcnt)
  if (tid < 32) {                         // wave 0 only: one TDM op per workgroup
    uint64_t ga = (uint64_t)(uintptr_t)src;
    uint32_t lds_base = (uint32_t)(uintptr_t)(void*)&rowbuf[0]; // flat addr[31:0] == LDS offset
    u32x4 g0;
    g0.x = 1u;                                            // count=1, user descriptor
    g0.y = lds_base;                                      // lds_addr [63:32]
    g0.z = (uint32_t)(ga & 0xffffffffull);                // global_addr lo
    g0.w = (uint32_t)((ga >> 32) & 0x01ffffffull)         // global_addr[56:32]
         | (2u << 30);                                    // type = 2 ("image")
    i32x8 g1;
    g1[0] = (int)(2u << 16);                              // data_size = 4 bytes; mask=0
    g1[1] = (int)(((uint32_t)M & 0xffffu) << 16);         // tensor_dim0 [15:0] -> bits 63:48
    g1[2] = (int)((((uint32_t)M >> 16) & 0xffffu)         // tensor_dim0 [31:16]
         | (1u << 16));                                   // tensor_dim1 = 1
    g1[3] = (int)(((uint32_t)M & 0xffffu) << 16);         // tile_dim0 = M
    g1[4] = 1;                                            // tile_dim1 = 1
    g1[5] = M;                                            // tensor_dim0_stride lo32
    g1[6] = 0;
    g1[7] = 0;
    i32x4 z4 = {0, 0, 0, 0};                              // groups 2/3 unused (2-D tensor)
#if defined(__clang_major__) && __clang_major__ >= 23
    i32x8 z8 = {0, 0, 0, 0, 0, 0, 0, 0};
    __builtin_amdgcn_tensor_load_to_lds(g0, g1, z4, z4, z8, 0);
#else
    __builtin_amdgcn_tensor_load_to_lds(g0, g1, z4, z4, 0);
#endif
    __builtin_amdgcn_s_wait_tensorcnt(0);
  }
#else
  for (int m = tid; m < M; m += BLOCK) rowbuf[m] = src[m];
#endif
  __syncthreads();

  // ---- Phase 1: value = sum_m eq_gate(m, addr) * |mem[row, m]| ----
  const float addrf = (float)addr[row];
  float partial = 0.0f;
  for (int m = tid; m < M; m += BLOCK) {
    float diff = (float)m - addrf;
    float w = sthr(diff + 0.5f) * sthr(0.5f - diff);
    partial += w * fabsf(rowbuf[m]);
  }

#ifdef USE_WMMA_RED
  // Block reduction as a dot product on the matrix pipe:
  // slot t = 64c + k*16 + m  <->  A-tile chunk c, element (M=m, K=k).
  // D = sum_c A_c x ones(4x16): every column of D holds rowsum(M); total = sum_M.
  red[tid] = partial;
  __syncthreads();
  if (wid == 0) {
    const int half = lane >> 4;      // A layout: lane L, VGPR r <-> K = r + 2*(L/16)
    const int l15  = lane & 15;      //                         M = L % 16
    v8f acc = {0.f, 0.f, 0.f, 0.f, 0.f, 0.f, 0.f, 0.f};
    v2f bones = {1.0f, 1.0f};        // ones B-matrix: layout-independent
    for (int c = 0; c < 4; ++c) {
      v2f a;
      a.x = red[64 * c + (2 * half + 0) * 16 + l15];
      a.y = red[64 * c + (2 * half + 1) * 16 + l15];
      acc = __builtin_amdgcn_wmma_f32_16x16x4_f32(false, a, false, bones,
                                                  (short)0, acc, false, false);
    }
    float s = acc[0] + acc[1] + acc[2] + acc[3] + acc[4] + acc[5] + acc[6] + acc[7];
    s += __shfl_xor(s, 16, 32);      // rows 0-7 (lanes 0-15) + rows 8-15 (lanes 16-31)
    if (lane == 0) sval = s;
  }
  __syncthreads();
#else
  {
    float ws = wave_sum32(partial);
    if (lane == 0) wred[wid] = ws;
    __syncthreads();
    if (tid == 0) {
      float v = 0.0f;
      for (int i = 0; i < BLOCK / 32; ++i) v += wred[i];  // fixed order: deterministic
      sval = v;
    }
    __syncthreads();
  }
#endif
  const float value = sval;

  if (tid < OUTC) tokens[tid] = (tid == OUTC - 1) ? value : 0.0f;

  // ---- Phase 2: digit enumeration (p=0: 1000 q's, p=1: 102 q's, block-parallel) ----
  float p0 = 0.0f;
  for (int k = 0; k < 4; ++k) {
    int q = tid + k * BLOCK;
    if (q <= 999) {
      float qf = (float)q;
      p0 += sthr(value - qf + 0.5f) * sthr(qf + 1.0f - value - 0.5f) * qf;
    }
  }
  float p1 = 0.0f;
  if (tid <= 101) {
    float qf = (float)tid;
    p1 = sthr(value - qf * 10.0f + 0.5f) * sthr((qf + 1.0f) * 10.0f - value - 0.5f) * qf;
  }

  float q0v = 0.0f, q1v = 0.0f;      // valid on thread 0 after reduction
#ifdef USE_WMMA_RED
  // Pack p0 partials into D-rows 0-7, p1 partials into D-rows 8-15 (512 slots, 8 chunks).
  {
    int m  = tid & 7;
    int k  = (tid >> 3) & 3;
    int c  = tid >> 5;
    int pos = 64 * c + k * 16 + m;
    red[pos]     = p0;
    red[pos + 8] = p1;
  }
  __syncthreads();
  if (wid == 0) {
    const int half = lane >> 4;
    const int l15  = lane & 15;
    v8f acc = {0.f, 0.f, 0.f, 0.f, 0.f, 0.f, 0.f, 0.f};
    v2f bones = {1.0f, 1.0f};
    for (int c = 0; c < 8; ++c) {
      v2f a;
      a.x = red[64 * c + (2 * half + 0) * 16 + l15];
      a.y = red[64 * c + (2 * half + 1) * 16 + l15];
      acc = __builtin_amdgcn_wmma_f32_16x16x4_f32(false, a, false, bones,
                                                  (short)0, acc, false, false);
    }
    float s = acc[0] + acc[1] + acc[2] + acc[3] + acc[4] + acc[5] + acc[6] + acc[7];
    float sHi = __shfl(s, 16, 32);   // lanes 16-31 carry rows 8-15 (= p1)
    q0v = s;                         // lanes 0-15 carry rows 0-7   (= p0)
    q1v = sHi;
  }
  __syncthreads();
#else
  {
    float r0 = wave_sum32(p0);
    float r1 = wave_sum32(p1);
    if (lane == 0) { wred[wid] = r0; wred[8 + wid] = r1; }
    __syncthreads();
    if (tid == 0) {
      for (int i = 0; i < 8; ++i) { q0v += wred[i]; q1v += wred[8 + i]; }
    }
  }
#endif

  if (tid == 0) {
    float quot[6];
    quot[0] = q0v; quot[1] = q1v;
    const float dtab[6] = {1.f, 10.f, 100.f, 1000.f, 10000.f, 100000.f};
    const int   qtab[6] = {1000, 102, 12, 3, 2, 2};
    for (int p = 2; p < 6; ++p) {                         // 19 terms total: serial is fine
      float d = dtab[p], s = 0.0f;
      for (int q = 0; q < qtab[p]; ++q) {
        float qf = (float)q;
        s += sthr(value - qf * d + 0.5f) * sthr((qf + 1.0f) * d - value - 0.5f) * qf;
      }
      quot[p] = s;
    }
    int digits[6];
    for (int p = 0; p < 6; ++p) {
      float dg = quot[p] - floorf(quot[p] / 10.0f) * 10.0f;   // mod 10, then floor
      digits[p] = (int)floorf(dg);
    }
    // count_digits
    float cnt = 1.0f;
    for (int i = 1; i < 6; ++i) cnt += sthr(value - dtab[i] + 0.5f);
    int n = (int)floorf(cnt);
    if (n < 0) n = 0;
    if (n > 6) n = 6;
    // token scatter: j < n -> digit(n-1-j), j == n -> newline; OOB dropped
    int optr = out_ptr[row];
    for (int j = 0; j <= n; ++j) {
      float tok;
      if (j < n) {
        int pos = n - 1 - j;
        pos = pos < 0 ? 0 : (pos > 5 ? 5 : pos);
        tok = 48.0f + (float)digits[pos];
      } else {
        tok = 10.0f;
      }
      int wp = optr + j;
      if (wp >= 0 && wp < OUTC - 1) tokens[wp] = tok;
    }
  }
  __syncthreads();

  if (tid < OUTC) out[(size_t)row * OUTC + tid] = tokens[tid];
}

extern "C" void kernel_launch(void* const* d_in, const int* in_sizes, int n_in,
                              void* d_out, int out_size, void* d_ws, size_t ws_size,
                              hipStream_t stream) {
  (void)n_in; (void)out_size; (void)d_ws; (void)ws_size;
  const float* memory  = (const float*)d_in[0];
  const int*   addr    = (const int*)d_in[1];
  const int*   out_ptr = (const int*)d_in[2];
  float*       out     = (float*)d_out;
  const int B = in_sizes[1];
  int M = (B > 0) ? (in_sizes[0] / B) : 0;
  if (M > MAXM) M = MAXM;
  c4_print_kernel<<<dim3(B), dim3(BLOCK), 0, stream>>>(memory, addr, out_ptr, out, M);
}